// GCN_65240553226643
// MI455X (gfx1250) — compile-verified
//
#include <hip/hip_runtime.h>

#define NN 100000
#define NE 1600000

typedef __attribute__((ext_vector_type(2))) float v2f;
typedef __attribute__((ext_vector_type(8))) float v8f;

// ---------------- degree / norm ----------------

__global__ void init_deg_kernel(float* __restrict__ deg) {
    int i = blockIdx.x * blockDim.x + threadIdx.x;
    if (i < NN) deg[i] = 1.0f;   // self-loop contributes 1
}

__global__ void count_deg_kernel(const long long* __restrict__ ei,
                                 float* __restrict__ deg) {
    int e = blockIdx.x * blockDim.x + threadIdx.x;
    if (e >= NE) return;
    int d = (int)ei[NE + e];     // dst row of edge_index
    unsafeAtomicAdd(&deg[d], 1.0f);
}

__global__ void finalize_dinv_kernel(float* __restrict__ deg) {
    int i = blockIdx.x * blockDim.x + threadIdx.x;
    if (i < NN) deg[i] = rsqrtf(deg[i]);   // deg >= 1 always
}

// int64 edge_index -> int32 src/dst + precomputed symmetric norm
__global__ void edge_prep_kernel(const long long* __restrict__ ei,
                                 const float* __restrict__ dinv,
                                 int* __restrict__ src32,
                                 int* __restrict__ dst32,
                                 float* __restrict__ nrm) {
    int e = blockIdx.x * blockDim.x + threadIdx.x;
    if (e >= NE) return;
    int s = (int)ei[e];
    int d = (int)ei[NE + e];
    src32[e] = s;
    dst32[e] = d;
    nrm[e] = dinv[s] * dinv[d];
}

// ---------------- dense transform: C[M,F] = A[M,K] @ W[K,F] via f32 WMMA ----------------
// One block per 16-row tile; wave w handles column tile w. W staged in LDS
// (padded stride F+8 dwords: half-waves read rows k / k+2 -> 2*S % 64 == 16
// banks apart -> conflict-free; stride*4 bytes stays 16B-aligned).
// A fragment (16x4 f32): lanes 0-15 -> M=lane, K {k,k+1}; lanes 16-31 -> K {k+2,k+3}.
// C/D fragment (16x16 f32, 8 VGPRs): vgpr v, lane l -> row v + 8*(l>>4), col l&15.

template <int K, int F, bool RELU>
__global__ __launch_bounds__(32 * (F / 16))
void gemm_wmma_kernel(const float* __restrict__ A, const float* __restrict__ W,
                      float* __restrict__ C) {
    constexpr int S  = F + 8;                 // padded LDS row stride (dwords)
    constexpr int NT = 32 * (F / 16);         // threads per block
    __shared__ float sW[K * S];

    // cooperative stage of W into LDS (float4 global loads, float4 LDS stores)
    constexpr int NV4 = K * F / 4;
#pragma unroll
    for (int i = threadIdx.x; i < NV4; i += NT) {
        int k  = i / (F / 4);
        int fc = i - k * (F / 4);
        float4 w4 = ((const float4*)W)[i];
        *(float4*)&sW[k * S + 4 * fc] = w4;
    }
    __syncthreads();

    const int lane  = threadIdx.x & 31;
    const int wave  = threadIdx.x >> 5;       // column tile index
    const int row0  = blockIdx.x << 4;
    const int col0  = wave << 4;
    const int mrow  = lane & 15;              // A row in tile / B,C column in tile
    const int khalf = (lane >> 4) << 1;       // 0 for lanes 0-15, 2 for lanes 16-31

    v8f acc = {};
    const float* ap = A + (size_t)(row0 + mrow) * K + khalf;
    const float* bp = &sW[khalf * S + col0 + mrow];

#pragma unroll
    for (int k = 0; k < K; k += 4) {
        v2f a = *(const v2f*)(ap + k);        // 8B aligned: khalf, k even
        if (RELU) { a.x = fmaxf(a.x, 0.0f); a.y = fmaxf(a.y, 0.0f); }
        v2f b;
        b.x = bp[k * S];
        b.y = bp[(k + 1) * S];
        acc = __builtin_amdgcn_wmma_f32_16x16x4_f32(
            /*neg_a=*/false, a, /*neg_b=*/false, b,
            /*c_mod=*/(short)0, acc, /*reuse_a=*/false, /*reuse_b=*/false);
    }

    const int rhi = (lane >> 4) << 3;         // 0 or 8
    float* cp = C + (size_t)row0 * F + col0 + mrow;
#pragma unroll
    for (int v = 0; v < 8; ++v)
        cp[(size_t)(v + rhi) * F] = acc[v];
}

// ---------------- aggregation ----------------
// out[i] = dinv[i]^2 * hw[i] + bias   (self-loop term; also zero-initializes out)

__global__ void agg_init_kernel(const float* __restrict__ hw,
                                const float* __restrict__ dinv,
                                const float* __restrict__ bias,
                                float* __restrict__ out, int fdiv4) {
    int idx = blockIdx.x * blockDim.x + threadIdx.x;
    int total = NN * fdiv4;
    if (idx >= total) return;
    int i = idx / fdiv4;
    int c = idx - i * fdiv4;
    float di = dinv[i];
    float s = di * di;
    float4 v = ((const float4*)hw)[idx];
    float4 b = ((const float4*)bias)[c];
    float4 o;
    o.x = fmaf(v.x, s, b.x);
    o.y = fmaf(v.y, s, b.y);
    o.z = fmaf(v.z, s, b.z);
    o.w = fmaf(v.w, s, b.w);
    ((float4*)out)[idx] = o;
}

// out[dst] += nrm[e] * hw[src]  -- one thread per (edge, float4 chunk)
template <int CH>   // CH = F/4 (32 or 16), power of two
__global__ void agg_edges_kernel(const int* __restrict__ src32,
                                 const int* __restrict__ dst32,
                                 const float* __restrict__ nrm,
                                 const float* __restrict__ hw,
                                 float* __restrict__ out) {
    constexpr int SH = (CH == 32) ? 5 : 4;
    int idx = blockIdx.x * blockDim.x + threadIdx.x;
    const int total = NE * CH;
    if (idx >= total) return;
    int e = idx >> SH;
    int c = idx & (CH - 1);
    int s = src32[e];
    int d = dst32[e];
    float n = nrm[e];
    float4 v = ((const float4*)hw)[s * CH + c];
    float* op = out + (size_t)d * (CH * 4) + c * 4;
    unsafeAtomicAdd(op + 0, v.x * n);
    unsafeAtomicAdd(op + 1, v.y * n);
    unsafeAtomicAdd(op + 2, v.z * n);
    unsafeAtomicAdd(op + 3, v.w * n);
}

// ---------------- launcher ----------------

extern "C" void kernel_launch(void* const* d_in, const int* in_sizes, int n_in,
                              void* d_out, int out_size, void* d_ws, size_t ws_size,
                              hipStream_t stream) {
    const float*     x  = (const float*)d_in[0];
    const long long* ei = (const long long*)d_in[1];
    const float*     W1 = (const float*)d_in[2];
    const float*     b1 = (const float*)d_in[3];
    const float*     W2 = (const float*)d_in[4];
    const float*     b2 = (const float*)d_in[5];
    const float*     W3 = (const float*)d_in[6];
    const float*     b3 = (const float*)d_in[7];
    float* out = (float*)d_out;

    // workspace layout (all 256B-aligned)
    char* ws = (char*)d_ws;
    size_t off = 0;
    auto alloc = [&](size_t bytes) {
        char* p = ws + off;
        off += (bytes + 255) & ~(size_t)255;
        return p;
    };
    float* dinv  = (float*)alloc((size_t)NN * 4);
    float* buf0  = (float*)alloc((size_t)NN * 128 * 4);   // hw (post-GEMM)
    float* buf1  = (float*)alloc((size_t)NN * 128 * 4);   // h  (post-aggregation)
    int*   src32 = (int*)  alloc((size_t)NE * 4);
    int*   dst32 = (int*)  alloc((size_t)NE * 4);
    float* nrm   = (float*)alloc((size_t)NE * 4);

    const int B = 256;

    // degree -> dinv -> per-edge norm
    init_deg_kernel<<<(NN + B - 1) / B, B, 0, stream>>>(dinv);
    count_deg_kernel<<<(NE + B - 1) / B, B, 0, stream>>>(ei, dinv);
    finalize_dinv_kernel<<<(NN + B - 1) / B, B, 0, stream>>>(dinv);
    edge_prep_kernel<<<(NE + B - 1) / B, B, 0, stream>>>(ei, dinv, src32, dst32, nrm);

    const int rowTiles = NN / 16;   // 6250, exact

    // layer 1: x[100k,64] @ W1[64,128] -> aggregate (+b1); relu fused into next GEMM
    gemm_wmma_kernel<64, 128, false><<<rowTiles, 256, 0, stream>>>(x, W1, buf0);
    agg_init_kernel<<<(NN * 32 + B - 1) / B, B, 0, stream>>>(buf0, dinv, b1, buf1, 32);
    agg_edges_kernel<32><<<(NE * 32 + B - 1) / B, B, 0, stream>>>(src32, dst32, nrm, buf0, buf1);

    // layer 2: relu(h1) @ W2[128,128]
    gemm_wmma_kernel<128, 128, true><<<rowTiles, 256, 0, stream>>>(buf1, W2, buf0);
    agg_init_kernel<<<(NN * 32 + B - 1) / B, B, 0, stream>>>(buf0, dinv, b2, buf1, 32);
    agg_edges_kernel<32><<<(NE * 32 + B - 1) / B, B, 0, stream>>>(src32, dst32, nrm, buf0, buf1);

    // layer 3: relu(h2) @ W3[128,64] -> final aggregate (+b3) into d_out
    gemm_wmma_kernel<128, 64, true><<<rowTiles, 128, 0, stream>>>(buf1, W3, buf0);
    agg_init_kernel<<<(NN * 16 + B - 1) / B, B, 0, stream>>>(buf0, dinv, b3, out, 16);
    agg_edges_kernel<16><<<(NE * 16 + B - 1) / B, B, 0, stream>>>(src32, dst32, nrm, buf0, out);
}